// EncoderLayer_28595892256986
// MI455X (gfx1250) — compile-verified
//
#include <hip/hip_runtime.h>
#include <hip/hip_bf16.h>

typedef float v2f __attribute__((ext_vector_type(2)));
typedef float v8f __attribute__((ext_vector_type(8)));

#define NDIM   65536
#define MDIM   32768
#define CDIM   4
#define DIN    64
#define SDIM   32
#define LDY    132   // padded LDS row stride (floats): 4*row spreads all 64 banks

// One block: 32 (b,m) pairs -> 128 rows for GEMM1/3, 32 rows for GEMM2.
// 8 waves:
//   waves 0..3 -> GEMM1 (merge): row tiles {w, w+4}, 4 col tiles, B-frags reused 2x
//   waves 4..7 -> GEMM2 (ms):    row tiles {0,1},   4 col tiles each, B reused 2x
//   barrier; all 8 waves -> GEMM3 (msamp): 2x2 tile block per wave (A & B reuse)
__global__ __launch_bounds__(256) void encoder_fused_wmma(
    const float* __restrict__ last, const float* __restrict__ sample,
    const int* __restrict__ child_l, const int* __restrict__ child_r,
    const int* __restrict__ child_s,
    const float* __restrict__ W_merge, const float* __restrict__ b_merge,
    const float* __restrict__ a_merge,
    const float* __restrict__ W_ms, const float* __restrict__ b_ms,
    const float* __restrict__ a_ms,
    const float* __restrict__ W_msamp, const float* __restrict__ b_msamp,
    const float* __restrict__ a_msamp,
    float* __restrict__ out)
{
    __shared__ float y[128 * LDY];   // concat(ans, s): 128 rows x 128 cols (padded)

    const int tid  = threadIdx.x;
    const int w    = tid >> 5;       // wave id 0..7
    const int lane = tid & 31;
    const int lo   = lane & 15;
    const int hi   = lane >> 4;      // half-wave select (K phase)
    const int p_base = blockIdx.x * 32;   // first (b,m) pair of this block

    if (w < 4) {
        // ------------- GEMM1: ans = prelu([lch|rch] @ W_merge + b) -------------
        const float* lb[2];
        const float* rb[2];
        #pragma unroll
        for (int i = 0; i < 2; ++i) {
            const int rt      = w + 4 * i;           // row tile 0..7
            const int r_block = rt * 16 + lo;        // row within block (0..127)
            const int p       = p_base + (r_block >> 2);
            const int c       = r_block & 3;
            const int b       = p >> 15;             // M = 2^15
            const int m       = p & (MDIM - 1);
            const int il = child_l[m];
            const int ir = child_r[m];
            lb[i] = last + (((size_t)(b * NDIM + il)) * CDIM + c) * DIN;
            rb[i] = last + (((size_t)(b * NDIM + ir)) * CDIM + c) * DIN;
        }

        v8f acc[2][4] = {};
        // K = 0..63 from left child
        for (int ks = 0; ks < 16; ++ks) {
            v2f a0 = *(const v2f*)(lb[0] + ks * 4 + 2 * hi);
            v2f a1 = *(const v2f*)(lb[1] + ks * 4 + 2 * hi);
            const int kb = ks * 4 + 2 * hi;
            #pragma unroll
            for (int ct = 0; ct < 4; ++ct) {
                v2f bb;
                bb.x = W_merge[(size_t)kb * 64 + ct * 16 + lo];
                bb.y = W_merge[(size_t)(kb + 1) * 64 + ct * 16 + lo];
                acc[0][ct] = __builtin_amdgcn_wmma_f32_16x16x4_f32(
                    false, a0, false, bb, (short)0, acc[0][ct], false, false);
                acc[1][ct] = __builtin_amdgcn_wmma_f32_16x16x4_f32(
                    false, a1, false, bb, (short)0, acc[1][ct], false, false);
            }
        }
        // K = 64..127 from right child
        for (int ks = 16; ks < 32; ++ks) {
            const int kk = ks * 4 - 64;
            v2f a0 = *(const v2f*)(rb[0] + kk + 2 * hi);
            v2f a1 = *(const v2f*)(rb[1] + kk + 2 * hi);
            const int kb = ks * 4 + 2 * hi;
            #pragma unroll
            for (int ct = 0; ct < 4; ++ct) {
                v2f bb;
                bb.x = W_merge[(size_t)kb * 64 + ct * 16 + lo];
                bb.y = W_merge[(size_t)(kb + 1) * 64 + ct * 16 + lo];
                acc[0][ct] = __builtin_amdgcn_wmma_f32_16x16x4_f32(
                    false, a0, false, bb, (short)0, acc[0][ct], false, false);
                acc[1][ct] = __builtin_amdgcn_wmma_f32_16x16x4_f32(
                    false, a1, false, bb, (short)0, acc[1][ct], false, false);
            }
        }
        const float am = a_merge[0];
        #pragma unroll
        for (int i = 0; i < 2; ++i) {
            const int rt = w + 4 * i;
            #pragma unroll
            for (int ct = 0; ct < 4; ++ct) {
                const float bias = b_merge[ct * 16 + lo];
                #pragma unroll
                for (int v = 0; v < 8; ++v) {
                    float val = acc[i][ct][v] + bias;
                    val = (val >= 0.0f) ? val : am * val;
                    const int Mrow = v + 8 * hi;
                    y[(rt * 16 + Mrow) * LDY + ct * 16 + lo] = val;  // ans -> y[:,0:64]
                }
            }
        }
    } else {
        // ------------- GEMM2: s = prelu(sample_flat @ W_ms + b) -------------
        const int wv = w - 4;
        const float* sb[2];
        #pragma unroll
        for (int q = 0; q < 2; ++q) {
            const int p = p_base + q * 16 + lo;
            const int b = p >> 15;
            const int m = p & (MDIM - 1);
            const int is = child_s[m];
            sb[q] = sample + ((size_t)(b * NDIM + is)) * (CDIM * SDIM);
        }

        v8f acc[2][4] = {};
        for (int ks = 0; ks < 32; ++ks) {
            v2f a0 = *(const v2f*)(sb[0] + ks * 4 + 2 * hi);
            v2f a1 = *(const v2f*)(sb[1] + ks * 4 + 2 * hi);
            const int kb = ks * 4 + 2 * hi;
            #pragma unroll
            for (int t = 0; t < 4; ++t) {
                const int ct2 = wv * 4 + t;                 // col tile 0..15 of 256
                v2f bb;
                bb.x = W_ms[(size_t)kb * 256 + ct2 * 16 + lo];
                bb.y = W_ms[(size_t)(kb + 1) * 256 + ct2 * 16 + lo];
                acc[0][t] = __builtin_amdgcn_wmma_f32_16x16x4_f32(
                    false, a0, false, bb, (short)0, acc[0][t], false, false);
                acc[1][t] = __builtin_amdgcn_wmma_f32_16x16x4_f32(
                    false, a1, false, bb, (short)0, acc[1][t], false, false);
            }
        }
        const float am = a_ms[0];
        #pragma unroll
        for (int q = 0; q < 2; ++q) {
            #pragma unroll
            for (int t = 0; t < 4; ++t) {
                const int ct2 = wv * 4 + t;
                const float bias = b_ms[ct2 * 16 + lo];
                #pragma unroll
                for (int v = 0; v < 8; ++v) {
                    float val = acc[q][t][v] + bias;
                    val = (val >= 0.0f) ? val : am * val;
                    const int prow = q * 16 + v + 8 * hi;   // local (b,m) row 0..31
                    const int n    = ct2 * 16 + lo;         // 0..255
                    const int yr   = prow * 4 + (n >> 6);   // reshape (32,256)->(128,64)
                    const int yc   = 64 + (n & 63);         // s -> y[:,64:128]
                    y[yr * LDY + yc] = val;
                }
            }
        }
    }

    __syncthreads();

    // ------------- GEMM3: out = prelu(y @ W_msamp + b) -------------
    // 32 tiles (8 rt x 4 ct); wave w -> rt {2*(w&3), +1} x ct {2*(w>>2), +1}
    const int rt0 = 2 * (w & 3);
    const int ct0 = 2 * (w >> 2);
    const float am3 = a_msamp[0];

    v8f acc3[2][2] = {};
    for (int ks = 0; ks < 32; ++ks) {
        v2f a0 = *(const v2f*)(&y[((rt0 + 0) * 16 + lo) * LDY + ks * 4 + 2 * hi]);
        v2f a1 = *(const v2f*)(&y[((rt0 + 1) * 16 + lo) * LDY + ks * 4 + 2 * hi]);
        const int kb = ks * 4 + 2 * hi;
        #pragma unroll
        for (int j = 0; j < 2; ++j) {
            const int ct = ct0 + j;
            v2f bb;
            bb.x = W_msamp[(size_t)kb * 64 + ct * 16 + lo];
            bb.y = W_msamp[(size_t)(kb + 1) * 64 + ct * 16 + lo];
            acc3[0][j] = __builtin_amdgcn_wmma_f32_16x16x4_f32(
                false, a0, false, bb, (short)0, acc3[0][j], false, false);
            acc3[1][j] = __builtin_amdgcn_wmma_f32_16x16x4_f32(
                false, a1, false, bb, (short)0, acc3[1][j], false, false);
        }
    }
    #pragma unroll
    for (int i = 0; i < 2; ++i) {
        const int rt = rt0 + i;
        #pragma unroll
        for (int j = 0; j < 2; ++j) {
            const int ct = ct0 + j;
            const float bias = b_msamp[ct * 16 + lo];
            #pragma unroll
            for (int v = 0; v < 8; ++v) {
                float val = acc3[i][j][v] + bias;
                val = (val >= 0.0f) ? val : am3 * val;
                const int r_glob = p_base * 4 + rt * 16 + v + 8 * hi;  // ((b*M+m)*C+c)
                out[(size_t)r_glob * 64 + ct * 16 + lo] = val;
            }
        }
    }
}

extern "C" void kernel_launch(void* const* d_in, const int* in_sizes, int n_in,
                              void* d_out, int out_size, void* d_ws, size_t ws_size,
                              hipStream_t stream) {
    const float* last     = (const float*)d_in[0];
    const float* sample   = (const float*)d_in[1];
    const int*   child_l  = (const int*)d_in[2];
    const int*   child_r  = (const int*)d_in[3];
    const int*   child_s  = (const int*)d_in[4];
    const float* W_merge  = (const float*)d_in[5];
    const float* b_merge  = (const float*)d_in[6];
    const float* a_merge  = (const float*)d_in[7];
    const float* W_ms     = (const float*)d_in[8];
    const float* b_ms     = (const float*)d_in[9];
    const float* a_ms     = (const float*)d_in[10];
    const float* W_msamp  = (const float*)d_in[11];
    const float* b_msamp  = (const float*)d_in[12];
    const float* a_msamp  = (const float*)d_in[13];
    float* out = (float*)d_out;

    const int P = 4 * MDIM;           // B*M = 131072 (b,m) pairs
    dim3 grid(P / 32);                // 4096 blocks, 32 pairs each
    dim3 block(256);                  // 8 waves
    encoder_fused_wmma<<<grid, block, 0, stream>>>(
        last, sample, child_l, child_r, child_s,
        W_merge, b_merge, a_merge, W_ms, b_ms, a_ms,
        W_msamp, b_msamp, a_msamp, out);
}